// CRFLayer_64381559767440
// MI455X (gfx1250) — compile-verified
//
#include <hip/hip_runtime.h>
#include <hip/hip_bf16.h>
#include <stdint.h>

// ---------------------------------------------------------------------------
// CRF forward (log-partition) + gold score, MI455X / gfx1250.
// Scaled-probability recurrence:
//   p_new[j] = exp(emit[b,t,j]) * sum_i exp(trans[j,i]) * p[i];  renorm, logZ += log2(s)
// Transition mix done with V_WMMA_F32_16X16X4_F32 (A = exp(trans), static;
// B = p over 16 batch columns; two K-chunks cover K=5).
// Feature tiles streamed Global->LDS with the Tensor Data Mover (TDM),
// double-buffered, synchronized with s_wait_tensorcnt + workgroup barriers.
// ---------------------------------------------------------------------------

typedef __attribute__((ext_vector_type(2))) float        v2f;
typedef __attribute__((ext_vector_type(8))) float        v8f;
typedef __attribute__((ext_vector_type(4))) unsigned int v4u;
typedef __attribute__((ext_vector_type(8))) int          v8i;
typedef __attribute__((ext_vector_type(4))) int          v4i;

#define CRF_BS   8192
#define CRF_T    512
#define CRF_K    5
#define CHUNK    16                   // timesteps per LDS tile
#define ROWS     64                   // batch rows per block (4 waves * 16)
#define RSTRIDE  (CHUNK * CRF_K)      // 80 floats per row per chunk
#define NCHUNK   (CRF_T / CHUNK)      // 32

#define LN2F     0.6931471805599453f

// Issue a TDM 2D tile load: ROWS rows x RSTRIDE floats, row stride T*K floats.
// D# packing per CDNA5 ISA.
// amdgpu-toolchain (clang-23) 6-arg builtin form:
//   (uint32x4 g0, int32x8 g1, int32x4 g2, int32x4 g3, int32x8 gx, i32 cpol)
__device__ __forceinline__ void tdm_load_tile(const float* gsrc, const float* lds_dst)
{
    unsigned long long ga = (unsigned long long)(uintptr_t)gsrc;
    unsigned ldso = (unsigned)(uintptr_t)lds_dst;   // low 32 bits of flat addr = LDS byte offset

    v4u g0;
    g0.x = 1u;                                                  // count=1 (valid descriptor)
    g0.y = ldso;                                                // lds_addr
    g0.z = (unsigned)(ga & 0xffffffffull);                      // global_addr[31:0]
    g0.w = (unsigned)((ga >> 32) & 0x1ffffffull) | (2u << 30);  // global_addr[56:32] | type=2

    const unsigned td0 = RSTRIDE;      // tensor_dim0 (== tile width, always in-bounds)
    const unsigned td1 = ROWS;         // tensor_dim1
    const unsigned tl0 = RSTRIDE;      // tile_dim0 (elements)
    const unsigned tl1 = ROWS;         // tile_dim1 (rows)
    const unsigned long long s0 = (unsigned long long)(CRF_T * CRF_K); // row stride (elements)

    v8i g1;
    g1[0] = (int)(2u << 16);                                    // wg_mask=0, data_size=2 (4B)
    g1[1] = (int)((td0 & 0xffffu) << 16);                       // tensor_dim0[15:0] @ [31:16]
    g1[2] = (int)((td0 >> 16) | ((td1 & 0xffffu) << 16));       // td0 hi | td1 lo
    g1[3] = (int)((td1 >> 16) | (tl0 << 16));                   // td1 hi | tile_dim0
    g1[4] = (int)(tl1 & 0xffffu);                               // tile_dim1 | tile_dim2=0
    g1[5] = (int)(s0 & 0xffffffffull);                          // tensor_dim0_stride[31:0]
    g1[6] = (int)((s0 >> 32) & 0xffffull);                      // stride0 hi | stride1 lo = 0
    g1[7] = 0;                                                  // stride1 hi = 0

    v4i z4 = {0, 0, 0, 0};                                      // groups 2/3: 2D tensor -> zero
    v8i z8 = {0, 0, 0, 0, 0, 0, 0, 0};                          // extended group: unused
    __builtin_amdgcn_tensor_load_to_lds(g0, g1, z4, z4, z8, 0);
}

__global__ __launch_bounds__(128)
void crf_forward_kernel(const float* __restrict__ feature,
                        const int*   __restrict__ tags,
                        const float* __restrict__ transitions,
                        float*       __restrict__ partial)
{
    __shared__ float tile[2][ROWS * RSTRIDE];   // double-buffered feature tiles (40 KB)
    __shared__ float trans_s[32];
    __shared__ float wsum[4];

    const int tid  = threadIdx.x;
    const int lane = tid & 31;
    const int wv   = tid >> 5;
    const int b0   = blockIdx.x * ROWS;

    if (tid < CRF_K * CRF_K) trans_s[tid] = transitions[tid];
    __syncthreads();

    // A = exp(transitions) in 16x4 f32 WMMA A layout (M=j target tag, K=i source tag).
    // Lanes 0-15: M=lane, VGPR0=K0, VGPR1=K1.  Lanes 16-31: M=lane-16, VGPR0=K2, VGPR1=K3.
    const int jrow  = lane & 15;
    const int ibase = (lane >> 4) * 2;   // 0 or 2
    v2f A1, A2;
    {
        float a0 = 0.f, a1 = 0.f, a2 = 0.f;
        if (jrow < CRF_K) {
            a0 = __expf(trans_s[jrow * CRF_K + ibase]);      // i = 0 or 2
            a1 = __expf(trans_s[jrow * CRF_K + ibase + 1]);  // i = 1 or 3
            if (lane < 16) a2 = __expf(trans_s[jrow * CRF_K + 4]); // i = 4 (chunk 2)
        }
        A1.x = a0; A1.y = a1;
        A2.x = a2; A2.y = 0.f;          // i = 5..7 are zero padding
    }

    // Prime the TDM pipeline: chunks 0 and 1 (wave 0 only; TENSORcnt is per-wave).
    const float* fbase = feature + (size_t)b0 * CRF_T * CRF_K;
    if (wv == 0) {
        tdm_load_tile(fbase, tile[0]);
        tdm_load_tile(fbase + RSTRIDE, tile[1]);
    }

    // B = p in 4x16 f32 WMMA B layout; init p = delta(START=3):
    //   lane n:    V0 = p[0]=0, V1 = p[1]=0
    //   lane n+16: V0 = p[2]=0, V1 = p[3]=1
    v2f B1, B2;
    B1.x = 0.f; B1.y = (lane >= 16) ? 1.f : 0.f;
    B2.x = 0.f; B2.y = 0.f;             // p[4]=0, padding

    const int  bl     = wv * 16 + (lane & 15);     // local batch row (lanes 0-15)
    const int  b      = b0 + bl;
    const int* tagrow = tags + (size_t)b * CRF_T;

    float logZ2 = 0.f, gold = 0.f;      // logZ accumulated in log2 units
    int   prev = 3;                     // START
    float q0 = 0.f, q1 = 0.f, q2 = 0.f, q3 = 0.f, q4 = 0.f;

    for (int c = 0; c < NCHUNK; ++c) {
        // Prefetch this chunk's 16 gold tags into registers (4x b128 loads);
        // removes the per-step dependent global load + wait from the hot loop.
        int tloc[CHUNK];
        if (lane < 16) {
            const int4* tp = (const int4*)(tagrow + c * CHUNK);
            #pragma unroll
            for (int i = 0; i < CHUNK / 4; ++i) {
                int4 v = tp[i];
                tloc[4 * i + 0] = v.x; tloc[4 * i + 1] = v.y;
                tloc[4 * i + 2] = v.z; tloc[4 * i + 3] = v.w;
            }
        }

        if (wv == 0) {
            if (c + 1 < NCHUNK) __builtin_amdgcn_s_wait_tensorcnt((short)1); // oldest done
            else                __builtin_amdgcn_s_wait_tensorcnt((short)0); // last chunk
        }
        __syncthreads();                 // tile[c&1] ready for all waves
        const float* tb = tile[c & 1];

        #pragma unroll
        for (int tl = 0; tl < CHUNK; ++tl) {
            // D[j, b16] = sum_i E[j,i] * p[i, b16]  (two K-chunks, chained accumulate)
            v8f d = {0.f, 0.f, 0.f, 0.f, 0.f, 0.f, 0.f, 0.f};
            d = __builtin_amdgcn_wmma_f32_16x16x4_f32(false, A1, false, B1,
                                                      (short)0, d, false, false);
            d = __builtin_amdgcn_wmma_f32_16x16x4_f32(false, A2, false, B2,
                                                      (short)0, d, false, false);
            if (lane < 16) {
                const float* em = tb + bl * RSTRIDE + tl * CRF_K;
                float e0 = __expf(em[0]), e1 = __expf(em[1]), e2 = __expf(em[2]);
                float e3 = __expf(em[3]), e4 = __expf(em[4]);
                q0 = d[0] * e0; q1 = d[1] * e1; q2 = d[2] * e2;
                q3 = d[3] * e3; q4 = d[4] * e4;
                float s = q0 + q1 + q2 + q3 + q4;
                // s is renormalized every step -> never denormal: raw v_log/v_rcp are safe.
                logZ2 += __builtin_amdgcn_logf(s);          // log2(s)
                float inv = __builtin_amdgcn_rcpf(s);       // ~1 ulp, cancels through logZ
                q0 *= inv; q1 *= inv; q2 *= inv; q3 *= inv; q4 *= inv;
                // gold path: trans[cur, prev] + emit[b, t, cur]
                int tc = tloc[tl];
                gold += trans_s[tc * CRF_K + prev] + em[tc];
                prev = tc;
            }
            // Rebuild B for next step: rows 2,3 move to the upper lane-half.
            const int sl = lane & 15;
            float u2 = __shfl(q2, sl, 32);
            float u3 = __shfl(q3, sl, 32);
            B1.x = (lane < 16) ? q0 : u2;
            B1.y = (lane < 16) ? q1 : u3;
            B2.x = (lane < 16) ? q4 : 0.f;
            B2.y = 0.f;
        }
        __syncthreads();                 // everyone done reading tile[c&1]
        if (wv == 0 && c + 2 < NCHUNK)
            tdm_load_tile(fbase + (size_t)(c + 2) * RSTRIDE, tile[c & 1]);
    }

    float res = 0.f;
    if (lane < 16) {
        // terminal: logsumexp_j(fwd[j] + trans[STOP, j]); exp(NEG) -> 0 handles trans[4,4]
        float term = q0 * __expf(trans_s[4 * CRF_K + 0])
                   + q1 * __expf(trans_s[4 * CRF_K + 1])
                   + q2 * __expf(trans_s[4 * CRF_K + 2])
                   + q3 * __expf(trans_s[4 * CRF_K + 3])
                   + q4 * __expf(trans_s[4 * CRF_K + 4]);
        float fwd = logZ2 * LN2F + __logf(term);
        gold += trans_s[4 * CRF_K + prev];        // trans[STOP, last_tag]
        res = fwd - gold;
    }
    // Deterministic fixed-tree wave reduction (upper lanes contribute 0).
    res += __shfl_down(res, 16, 32);
    res += __shfl_down(res, 8, 32);
    res += __shfl_down(res, 4, 32);
    res += __shfl_down(res, 2, 32);
    res += __shfl_down(res, 1, 32);
    if (lane == 0) wsum[wv] = res;
    __syncthreads();
    if (tid == 0) partial[blockIdx.x] = (wsum[0] + wsum[1]) + (wsum[2] + wsum[3]);
}

__global__ __launch_bounds__(128)
void crf_reduce_kernel(const float* __restrict__ partial, float* __restrict__ out, int n)
{
    __shared__ float s[128];
    const int t = threadIdx.x;
    s[t] = (t < n) ? partial[t] : 0.f;
    __syncthreads();
    for (int off = 64; off > 0; off >>= 1) {
        if (t < off) s[t] += s[t + off];
        __syncthreads();
    }
    if (t == 0) out[0] = s[0];
}

extern "C" void kernel_launch(void* const* d_in, const int* in_sizes, int n_in,
                              void* d_out, int out_size, void* d_ws, size_t ws_size,
                              hipStream_t stream)
{
    const float* feature = (const float*)d_in[0];   // [BS, T, K] f32
    const int*   tags    = (const int*)d_in[1];     // [BS, T] i32
    const float* trans   = (const float*)d_in[2];   // [K, K] f32
    float*       out     = (float*)d_out;           // scalar f32
    float*       partial = (float*)d_ws;            // >= 128 floats

    const int nblocks = CRF_BS / ROWS;              // 128
    crf_forward_kernel<<<nblocks, 128, 0, stream>>>(feature, tags, trans, partial);
    crf_reduce_kernel<<<1, 128, 0, stream>>>(partial, out, nblocks);

    (void)in_sizes; (void)n_in; (void)out_size; (void)ws_size;
}